// RobustGraphConvolution_11570641896115
// MI455X (gfx1250) — compile-verified
//
#include <hip/hip_runtime.h>
#include <cstdint>

// RobustGraphConvolution for MI455X (gfx1250, wave32)
// One wave32 per node: gather 64 neighbor rows -> LDS, fp32 WMMA gram matrix,
// L2 distances via norm trick, softmax over 32 candidates, weighted output.

typedef float v2f __attribute__((ext_vector_type(2)));
typedef float v8f __attribute__((ext_vector_type(8)));

#define DEG    64
#define KCAND  32
#define DFEAT  64
#define ST     66            // padded LDS row stride (floats): breaks 64-bank alignment
#define EPSV   1.1920929e-5f // 100 * FLT_EPSILON

__global__ __launch_bounds__(32)
void rgc_kernel(const float* __restrict__ x,
                const long long* __restrict__ neigh,
                float* __restrict__ out,
                int n_nodes)
{
    __shared__ float lds_xn[DEG * ST];   // gathered neighbor rows (xc = rows 0..31)
    __shared__ float lds_nn[DEG];        // squared norms per gathered row
    __shared__ float lds_dist[KCAND];    // per-candidate summed distances

    const int node = blockIdx.x;
    if (node >= n_nodes) return;         // block-uniform; EXEC stays all-ones inside
    const int lane = threadIdx.x;        // 0..31 (wave32)

    // ---------------- gather: 64 rows of 64 floats into LDS ----------------
    const long long base = (long long)node * DEG;
    long long i0 = neigh[base + lane];
    long long i1 = neigh[base + lane + 32];
    for (int r = 0; r < DEG; ++r) {
        long long nb = (r < 32) ? __shfl(i0, r, 32) : __shfl(i1, r - 32, 32);
        const float2* src = reinterpret_cast<const float2*>(x + nb * DFEAT);
        float2 v = src[lane];                              // lane covers cols 2l,2l+1
        *reinterpret_cast<float2*>(&lds_xn[r * ST + 2 * lane]) = v;
    }
    __syncthreads();

    // ---------------- squared row norms (staggered: conflict-free) ----------
    {
        float s0 = 0.f, s1 = 0.f;
        const int m0 = lane, m1 = lane + 32;
        #pragma unroll
        for (int dd = 0; dd < DFEAT; ++dd) {
            int d = (lane + dd) & 63;
            float a = lds_xn[m0 * ST + d]; s0 += a * a;
            float b = lds_xn[m1 * ST + d]; s1 += b * b;
        }
        lds_nn[m0] = s0;
        lds_nn[m1] = s1;
    }
    __syncthreads();

    // ---------------- fp32 WMMA gram matrix: G = Xc(32x64) * Xn^T(64x64) ----
    // A 16x4 f32 frag layout == B 4x16 frag layout (transposed): both are
    // lds[(tile*16 + lane%16)*ST + ks*4 + (lane>>4)*2 .. +1]
    const int rsub  = lane & 15;
    const int khalf = lane >> 4;

    v8f acc[2][4];
    #pragma unroll
    for (int mt = 0; mt < 2; ++mt)
        #pragma unroll
        for (int nt = 0; nt < 4; ++nt)
            acc[mt][nt] = (v8f){0.f,0.f,0.f,0.f,0.f,0.f,0.f,0.f};

    for (int ks = 0; ks < 16; ++ks) {
        const int col = ks * 4 + khalf * 2;
        v2f afrag[2], bfrag[4];
        #pragma unroll
        for (int mt = 0; mt < 2; ++mt)
            afrag[mt] = *reinterpret_cast<const v2f*>(&lds_xn[(mt * 16 + rsub) * ST + col]);
        #pragma unroll
        for (int nt = 0; nt < 4; ++nt)
            bfrag[nt] = *reinterpret_cast<const v2f*>(&lds_xn[(nt * 16 + rsub) * ST + col]);

        #pragma unroll
        for (int mt = 0; mt < 2; ++mt)
            #pragma unroll
            for (int nt = 0; nt < 4; ++nt)
                acc[mt][nt] = __builtin_amdgcn_wmma_f32_16x16x4_f32(
                    /*neg_a=*/false, afrag[mt],
                    /*neg_b=*/false, bfrag[nt],
                    /*c_mod=*/(short)0, acc[mt][nt],
                    /*reuse_a=*/false, /*reuse_b=*/false);
    }

    // ---------------- distances: sq = |nc[k] + nn[m] - 2G| ; dist[k] = sum_m sqrt(sq+eps)
    // C/D layout: VGPR v, lanes0-15 -> row mt*16+v (cols nt*16+lane), lanes16-31 -> row +8
    #pragma unroll
    for (int mt = 0; mt < 2; ++mt) {
        #pragma unroll
        for (int v = 0; v < 8; ++v) {
            const int row = mt * 16 + v + 8 * khalf;   // candidate index k (0..31)
            const float nc = lds_nn[row];
            float dsum = 0.f;
            #pragma unroll
            for (int nt = 0; nt < 4; ++nt) {
                const int m = nt * 16 + rsub;
                float sq = nc + lds_nn[m] - 2.0f * acc[mt][nt][v];
                dsum += sqrtf(fabsf(sq) + EPSV);
            }
            // reduce across the 16-lane group (xor masks < 16 stay in group)
            #pragma unroll
            for (int off = 1; off < 16; off <<= 1)
                dsum += __shfl_xor(dsum, off, 32);
            if (rsub == 0) lds_dist[row] = dsum;       // lanes 0 and 16: unique rows
        }
    }
    __syncthreads();

    // ---------------- softmax over 32 candidates (lane k = lane) ------------
    float logit = -lds_dist[lane];                     // TEMPERATURE = 1
    float lmax = logit;
    #pragma unroll
    for (int off = 16; off > 0; off >>= 1)
        lmax = fmaxf(lmax, __shfl_xor(lmax, off, 32));
    float e = __expf(logit - lmax);
    float esum = e;
    #pragma unroll
    for (int off = 16; off > 0; off >>= 1)
        esum += __shfl_xor(esum, off, 32);
    const float w = e / esum;

    // ---------------- output: out[n] = DEG * sum_k w[k] * xc[k] -------------
    float acc0 = 0.f, acc1 = 0.f;                      // lane handles d=lane, d=lane+32
    #pragma unroll
    for (int k = 0; k < KCAND; ++k) {
        float wk = __shfl(w, k, 32);
        acc0 += wk * lds_xn[k * ST + lane];            // bank = (2k+lane)%64: conflict-free
        acc1 += wk * lds_xn[k * ST + lane + 32];
    }
    float* dst = out + (long long)node * DFEAT;
    dst[lane]      = 64.0f * acc0;                     // row_sum = DEG = 64
    dst[lane + 32] = 64.0f * acc1;
}

extern "C" void kernel_launch(void* const* d_in, const int* in_sizes, int n_in,
                              void* d_out, int out_size, void* d_ws, size_t ws_size,
                              hipStream_t stream) {
    const float*     x     = (const float*)d_in[0];
    const long long* neigh = (const long long*)d_in[1];
    float*           out   = (float*)d_out;
    const int n_nodes = in_sizes[0] / DFEAT;           // x is (N, 64)
    rgc_kernel<<<n_nodes, 32, 0, stream>>>(x, neigh, out, n_nodes);
}